// VoronoiPropagation_5093831213733
// MI455X (gfx1250) — compile-verified
//
#include <hip/hip_runtime.h>
#include <stdint.h>

// ---------------- problem constants (match reference) ----------------
#define BATCH     64
#define H         224
#define W         224
#define NCLUST    196
#define GRIDC     14          // 14x14 centroid grid, cell 16x16 -> (16i+8, 16j+8)
#define NEIGH     10
#define NUM_ITERS 50
#define GRAD_W    10.0f
#define COLOR_W   10.0f

#define TPB  256              // 8 wave32 waves per block
#define PXT  4                // pixels per thread (16B vector traffic)
#define TILE (TPB * PXT)      // 1024 pixels / 4KB per block

#define PLANE ((size_t)H * (size_t)W)          // 50176
#define PTOT  ((size_t)BATCH * PLANE)          // 3,211,264 (divisible by 1024)
#define OCCW  ((int)((PLANE + 31) / 32))       // 1568 words of occupancy bits

typedef uint32_t u32x4 __attribute__((ext_vector_type(4)));
typedef uint32_t u32x8 __attribute__((ext_vector_type(8)));

// ---------------- CDNA5 async global->LDS helpers ----------------
// Per-lane 16B async copy global->LDS, tracked by ASYNCcnt (CDNA5 ISA).
// LDS destination = low 32 bits of the flat shared pointer (LDS aperture truncates
// addr[31:0] to the LDS byte offset, ISA §10.2).
__device__ __forceinline__ void async_g2l_b128(void* lds_generic, const void* gptr) {
  uint32_t ldsoff = (uint32_t)(uintptr_t)lds_generic;
  asm volatile("global_load_async_to_lds_b128 %0, %1, off"
               :: "v"(ldsoff), "v"((uint64_t)(uintptr_t)gptr)
               : "memory");
}
__device__ __forceinline__ void wait_async0() {
  asm volatile("s_wait_asynccnt 0x0" ::: "memory");
}

// Tensor Data Mover: 1D contiguous tile (nelem f32) global->LDS via a D#
// descriptor (CDNA5 ISA). Group0: count=1, lds_addr, global_addr,
// type=2. Group1: data_size=4B, tensor_dim0=tile_dim0=nelem, no multicast/
// pad/iterate. 2-group form: VADDR2/VADDR3 omitted (<=2D tensor).
// Tracked by TENSORcnt; EXEC is ignored by TDM ops, so issue from one wave.
__device__ __forceinline__ void tdm_load_1d(void* lds_generic, const void* gptr,
                                            uint32_t nelem) {
  uint64_t ga = (uint64_t)(uintptr_t)gptr;
  u32x4 g0;
  g0[0] = 1u;                                                  // count=1 (valid D#)
  g0[1] = (uint32_t)(uintptr_t)lds_generic;                    // lds_addr (bytes)
  g0[2] = (uint32_t)ga;                                        // global_addr[31:0]
  g0[3] = (uint32_t)((ga >> 32) & 0x01FFFFFFu) | (2u << 30);   // addr[56:32] | type=2
  u32x8 g1;
  g1[0] = (2u << 16);                         // workgroup_mask=0, data_size=2 (4B)
  g1[1] = (nelem & 0xFFFFu) << 16;            // tensor_dim0[15:0]
  g1[2] = (nelem >> 16) & 0xFFFFu;            // tensor_dim0[31:16], tensor_dim1=0
  g1[3] = (nelem & 0xFFFFu) << 16;            // tile_dim0 = nelem (16-bit)
  g1[4] = 0u; g1[5] = 0u; g1[6] = 0u; g1[7] = 0u;
  asm volatile("tensor_load_to_lds %0, %1" :: "s"(g0), "s"(g1) : "memory");
}
__device__ __forceinline__ void wait_tensor0() {
#if __has_builtin(__builtin_amdgcn_s_wait_tensorcnt)
  __builtin_amdgcn_s_wait_tensorcnt(0);
#else
  asm volatile("s_wait_tensorcnt 0x0" ::: "memory");
#endif
}

// ---------------- K2 (hot): one directional relaxation pass ----------------
// out[h,w] = min(in[h,w], in[(h-dy)%H,(w-dx)%W] + cost_d[h,w]); mask follows winner.
// Cost tile (4KB/block) arrives via one TDM tensor_load_to_lds; self dist/mask
// arrive via per-lane async b128 loads; gathered neighbor lines are warmed with
// global_prefetch_b8 and loaded normally while the DMA runs.
__global__ void vp_pass(const float* __restrict__ din, const int* __restrict__ min_,
                        float* __restrict__ dout, int* __restrict__ mout,
                        const float* __restrict__ cost, int dy, int dx) {
  __shared__ __align__(16) float s_dist[TILE];
  __shared__ __align__(16) float s_cost[TILE];
  __shared__ __align__(16) int   s_mask[TILE];

  const int t = threadIdx.x;
  const size_t tile0 = (size_t)blockIdx.x * TILE;
  const size_t idx0  = tile0 + (size_t)t * PXT;

  // One TDM descriptor per block DMAs the whole cost tile (TENSORcnt += 1).
  if (t == 0) tdm_load_1d(s_cost, cost + tile0, TILE);

  // Per-lane async staging of self dist / self mask (ASYNCcnt += 2 per wave).
  async_g2l_b128(&s_dist[t * PXT], din  + idx0);
  async_g2l_b128(&s_mask[t * PXT], min_ + idx0);

  // Speculatively warm the neighbor cachelines (global_prefetch_b8).
  long an = (long)idx0 - ((long)dy * W + (long)dx);
  if (an < 0) an = 0;
  if (an > (long)(PTOT - PXT)) an = (long)(PTOT - PXT);
  __builtin_prefetch(din  + an, 0, 3);
  __builtin_prefetch(min_ + an, 0, 3);

  // Gather the (wrapped) neighbor dist/mask with regular loads while DMA runs.
  float nd[PXT]; int nm[PXT];
#pragma unroll
  for (int i = 0; i < PXT; ++i) {
    size_t idx = idx0 + i;
    int wq = (int)(idx % W);
    int hq = (int)((idx / W) % H);
    int b  = (int)(idx / PLANE);
    int nh = hq - dy; nh += (nh < 0) ? H : 0; nh -= (nh >= H) ? H : 0;
    int nw = wq - dx; nw += (nw < 0) ? W : 0; nw -= (nw >= W) ? W : 0;
    size_t nidx = ((size_t)b * H + nh) * W + nw;
    nd[i] = din[nidx];
    nm[i] = min_[nidx];
  }

  wait_async0();                      // own-wave LDS writes visible
  if (t == 0) wait_tensor0();         // TDM tile landed
  __syncthreads();                    // publish s_cost to all waves

  float od[PXT]; int om[PXT];
#pragma unroll
  for (int i = 0; i < PXT; ++i) {
    float self = s_dist[t * PXT + i];
    float cand = nd[i] + s_cost[t * PXT + i];
    bool u = cand < self;
    od[i] = u ? cand : self;
    om[i] = u ? nm[i] : s_mask[t * PXT + i];
  }
  *(float4*)(dout + idx0) = make_float4(od[0], od[1], od[2], od[3]);
  *(int4*)(mout + idx0)   = make_int4(om[0], om[1], om[2], om[3]);
}

// ---------------- K0: cost precompute + dist/mask init ----------------
// cost[d][b,h,w] = grad^4*GRAD_W + COLOR_W * sum_c |x[c,h,w] - x[c,(h-dy)%H,(w-dx)%W]|
// (jnp.roll wraps). dist=+inf, mask=-1.
__global__ void vp_precompute(const float* __restrict__ x, const float* __restrict__ grad,
                              float* __restrict__ dist, int* __restrict__ mask,
                              float* __restrict__ cost) {
  size_t idx = (size_t)blockIdx.x * TPB + threadIdx.x;
  if (idx >= PTOT) return;
  int wq = (int)(idx % W);
  int hq = (int)((idx / W) % H);
  int b  = (int)(idx / PLANE);

  float g  = grad[idx];
  float g2 = g * g;
  float wg = g2 * g2 * GRAD_W;              // grad ** 4.0 * GRAD_W

  size_t xb = (size_t)b * 3 * PLANE;
  size_t p  = (size_t)hq * W + wq;
  float c0 = x[xb + p];
  float c1 = x[xb + PLANE + p];
  float c2 = x[xb + 2 * PLANE + p];

  const int dys[4] = {-1, 1, 0, 0};
  const int dxs[4] = {0, 0, -1, 1};
#pragma unroll
  for (int d = 0; d < 4; ++d) {
    int nh = hq - dys[d]; nh += (nh < 0) ? H : 0; nh -= (nh >= H) ? H : 0;
    int nw = wq - dxs[d]; nw += (nw < 0) ? W : 0; nw -= (nw >= W) ? W : 0;
    size_t np = (size_t)nh * W + nw;
    float pen = fabsf(c0 - x[xb + np]) +
                fabsf(c1 - x[xb + PLANE + np]) +
                fabsf(c2 - x[xb + 2 * PLANE + np]);
    cost[(size_t)d * PTOT + idx] = wg + pen * COLOR_W;
  }
  dist[idx] = __builtin_inff();
  mask[idx] = -1;
}

// ---------------- K1: sequential centroid snapping (one block per image) ----------
// Replicates find_nearest_minima: per centroid (in order), window min over the
// 20x20 neighborhood, first row-major unoccupied cell equal to the min; occupancy
// dedup across centroids. grad is uniform[0,1) so float order == uint bit order.
__global__ void vp_seed(const float* __restrict__ grad, float* __restrict__ dist,
                        int* __restrict__ mask, int* __restrict__ cents) {
  __shared__ unsigned occ[OCCW];
  __shared__ unsigned s_min;
  __shared__ unsigned s_idx;

  const int b = blockIdx.x;
  const int t = threadIdx.x;
  const float* g = grad + (size_t)b * PLANE;

  for (int i = t; i < OCCW; i += TPB) occ[i] = 0u;
  __syncthreads();

  for (int k = 0; k < NCLUST; ++k) {
    const int cy = 16 * (k / GRIDC) + 8;
    const int cx = 16 * (k % GRIDC) + 8;
    const int ymin = (cy - NEIGH < 0) ? 0 : cy - NEIGH;
    const int ymax = (cy + NEIGH > H) ? H : cy + NEIGH;
    const int xmin = (cx - NEIGH < 0) ? 0 : cx - NEIGH;
    const int xmax = (cx + NEIGH > W) ? W : cx + NEIGH;
    const int ww = xmax - xmin;
    const int n  = (ymax - ymin) * ww;

    if (t == 0) { s_min = 0x7f800000u; s_idx = 0xffffffffu; }
    __syncthreads();

    for (int e = t; e < n; e += TPB) {
      int yy = ymin + e / ww, xx = xmin + e % ww;
      atomicMin(&s_min, __float_as_uint(g[yy * W + xx]));
    }
    __syncthreads();

    const unsigned mv = s_min;
    for (int e = t; e < n; e += TPB) {
      int yy = ymin + e / ww, xx = xmin + e % ww;
      unsigned flat = (unsigned)(yy * W + xx);
      if (__float_as_uint(g[flat]) == mv && !((occ[flat >> 5] >> (flat & 31)) & 1u))
        atomicMin(&s_idx, flat);
    }
    __syncthreads();

    if (t == 0) {
      int y = cy, xq = cx;
      unsigned fi = s_idx;
      if (fi != 0xffffffffu) {
        y = (int)(fi / W); xq = (int)(fi % W);
        occ[fi >> 5] |= 1u << (fi & 31);
      }
      cents[((size_t)b * NCLUST + k) * 2 + 0] = y;
      cents[((size_t)b * NCLUST + k) * 2 + 1] = xq;
      dist[(size_t)b * PLANE + (size_t)y * W + xq] = 0.0f;   // seed, order = k (last wins)
      mask[(size_t)b * PLANE + (size_t)y * W + xq] = k;
    }
    __syncthreads();
  }
}

// ---------------- K3: emit final mask into d_out ----------------
__global__ void vp_copy(const int* __restrict__ src, int* __restrict__ dst) {
  size_t i = ((size_t)blockIdx.x * TPB + threadIdx.x) * 4;
  if (i + 3 < PTOT + 4) *(int4*)(dst + i) = *(const int4*)(src + i);
}

// ---------------- host-side orchestration ----------------
extern "C" void kernel_launch(void* const* d_in, const int* in_sizes, int n_in,
                              void* d_out, int out_size, void* d_ws, size_t ws_size,
                              hipStream_t stream) {
  (void)in_sizes; (void)n_in; (void)out_size; (void)ws_size;
  const float* x    = (const float*)d_in[0];   // [64,3,224,224]
  const float* grad = (const float*)d_in[1];   // [64,224,224]
  int* out = (int*)d_out;                      // cents [64,196,2] i32 ++ mask [64,224,224] i32

  // Workspace: dist A/B, mask A/B, 4 cost planes (~98 MB total, L2-resident).
  char* ws = (char*)d_ws;
  float* distA = (float*)ws;  ws += PTOT * sizeof(float);
  float* distB = (float*)ws;  ws += PTOT * sizeof(float);
  int*   maskA = (int*)ws;    ws += PTOT * sizeof(int);
  int*   maskB = (int*)ws;    ws += PTOT * sizeof(int);
  float* cost  = (float*)ws;  // 4 * PTOT floats

  vp_precompute<<<(unsigned)(PTOT / TPB), TPB, 0, stream>>>(x, grad, distA, maskA, cost);
  vp_seed<<<BATCH, TPB, 0, stream>>>(grad, distA, maskA, out);

  static const int dys[4] = {-1, 1, 0, 0};
  static const int dxs[4] = {0, 0, -1, 1};
  float* rd = distA; int* rm = maskA;
  float* wd = distB; int* wm = maskB;
  for (int it = 0; it < NUM_ITERS; ++it) {
    for (int d = 0; d < 4; ++d) {            // Gauss-Seidel across dirs = one launch each
      vp_pass<<<(unsigned)(PTOT / TILE), TPB, 0, stream>>>(
          rd, rm, wd, wm, cost + (size_t)d * PTOT, dys[d], dxs[d]);
      float* td = rd; rd = wd; wd = td;
      int*   tm = rm; rm = wm; wm = tm;
    }
  }
  // 200 passes (even) -> final result lives in the A buffers (rm == maskA).
  vp_copy<<<(unsigned)(PTOT / (TPB * 4)), TPB, 0, stream>>>(
      rm, out + (size_t)BATCH * NCLUST * 2);
}